// SimpleGCN_59407987638622
// MI455X (gfx1250) — compile-verified
//
#include <hip/hip_runtime.h>
#include <hip/hip_bf16.h>

typedef __attribute__((ext_vector_type(2))) float v2f;
typedef __attribute__((ext_vector_type(8))) float v8f;

#define N_NODES 100000
#define N_EDGES 1600000
#define F_IN    500
#define HID     128
#define N_CLS   40

// ---------------- degree / normalization ----------------

__global__ __launch_bounds__(256) void deg_init_kernel(float* __restrict__ deg) {
    int i = blockIdx.x * blockDim.x + threadIdx.x;
    if (i < N_NODES) deg[i] = 1.0f;               // self-loop contribution
}

__global__ __launch_bounds__(256) void deg_count_kernel(const int* __restrict__ dst,
                                                        float* __restrict__ deg) {
    int e = blockIdx.x * blockDim.x + threadIdx.x;
    if (e < N_EDGES) unsafeAtomicAdd(&deg[dst[e]], 1.0f);
}

__global__ __launch_bounds__(256) void dinv_kernel(float* __restrict__ d) {
    int i = blockIdx.x * blockDim.x + threadIdx.x;
    if (i < N_NODES) d[i] = rsqrtf(d[i]);         // deg >= 1 always
}

// ---------------- W pre-pack into WMMA B-fragment layout ----------------
// Wp[(ks*NT + t)*32 + lane] = { W[ks*4+kk][col], W[ks*4+kk+1][col] },
// kk = (lane>>4)*2, col = t*16 + (lane&15); zero-padded for col >= Nc.

template<int NT>
__global__ __launch_bounds__(256) void pack_w_kernel(const float* __restrict__ W,
                                                     v2f* __restrict__ Wp,
                                                     int K, int Nc) {
    int idx = blockIdx.x * blockDim.x + threadIdx.x;
    int total = (K / 4) * NT * 32;
    if (idx >= total) return;
    int lane = idx & 31;
    int t    = (idx >> 5) % NT;
    int ks   = idx / (32 * NT);
    int col  = t * 16 + (lane & 15);
    int k    = ks * 4 + ((lane >> 4) * 2);
    v2f v;
    v.x = (col < Nc) ? W[(size_t)k * Nc + col]       : 0.0f;
    v.y = (col < Nc) ? W[(size_t)(k + 1) * Nc + col] : 0.0f;
    Wp[idx] = v;
}

// ---------------- f32 WMMA GEMM: Out[M,NC] = (relu?)A[M,K] @ W ----------------
// One wave computes a 32 x (16*NT) tile (two 16-row strips sharing B fragments):
// per k-step: 2 A b64 loads + NT B b64 loads + 2*NT WMMAs. Halves B L2 traffic.

template<int NT, int NC, bool RELU>
__global__ __launch_bounds__(128) void gemm_wmma_kernel(const float* __restrict__ A,
                                                        const v2f* __restrict__ Wp,
                                                        float* __restrict__ Out,
                                                        int M, int K) {
    const int wave = threadIdx.x >> 5;
    const int lane = threadIdx.x & 31;
    const int pair = blockIdx.x * 4 + wave;        // 32-row tile index
    if (pair * 32 >= M) return;

    const int kk  = (lane >> 4) * 2;               // K sub-offset per half-wave
    const int col = lane & 15;

    const float* __restrict__ Arow0 = A + (size_t)(pair * 32 + col) * K + kk;
    const float* __restrict__ Arow1 = Arow0 + (size_t)16 * K;
    const v2f*   __restrict__ wp    = Wp + lane;

    v8f acc0[NT], acc1[NT];
#pragma unroll
    for (int t = 0; t < NT; ++t) {
        acc0[t] = (v8f){0.f,0.f,0.f,0.f,0.f,0.f,0.f,0.f};
        acc1[t] = (v8f){0.f,0.f,0.f,0.f,0.f,0.f,0.f,0.f};
    }

    const int nsteps = K >> 2;
#pragma unroll 2
    for (int ks = 0; ks < nsteps; ++ks) {
        v2f a0 = *reinterpret_cast<const v2f*>(Arow0 + ks * 4);  // global_load_b64
        v2f a1 = *reinterpret_cast<const v2f*>(Arow1 + ks * 4);  // global_load_b64
        if (RELU) {
            a0.x = fmaxf(a0.x, 0.0f); a0.y = fmaxf(a0.y, 0.0f);
            a1.x = fmaxf(a1.x, 0.0f); a1.y = fmaxf(a1.y, 0.0f);
        }
#pragma unroll
        for (int t = 0; t < NT; ++t) {
            v2f b = wp[(ks * NT + t) * 32];                      // global_load_b64
            acc0[t] = __builtin_amdgcn_wmma_f32_16x16x4_f32(
                          false, a0, false, b, (short)0, acc0[t], false, false);
            acc1[t] = __builtin_amdgcn_wmma_f32_16x16x4_f32(
                          false, a1, false, b, (short)0, acc1[t], false, false);
        }
    }

    const int rbase0 = pair * 32 + ((lane < 16) ? 0 : 8);
    const int rbase1 = rbase0 + 16;
#pragma unroll
    for (int t = 0; t < NT; ++t) {
        int c = t * 16 + col;
        if ((NC % 16 == 0) || (c < NC)) {          // compile-time for NC=128
#pragma unroll
            for (int r = 0; r < 8; ++r) {
                Out[(size_t)(rbase0 + r) * NC + c] = acc0[t][r];
                Out[(size_t)(rbase1 + r) * NC + c] = acc1[t][r];
            }
        }
    }
}

// ---------------- aggregation init: Y = b + dinv_i^2 * h_i (float4) ----------------

template<int F>
__global__ __launch_bounds__(256) void self_bias_kernel(const float* __restrict__ h,
                                                        const float* __restrict__ dinv,
                                                        const float* __restrict__ bias,
                                                        float* __restrict__ out,
                                                        int n) {
    constexpr int F4 = F / 4;
    long idx = (long)blockIdx.x * blockDim.x + threadIdx.x;
    if (idx >= (long)n * F4) return;
    int i  = (int)(idx / F4);
    int f4 = (int)(idx % F4);
    float di = dinv[i];
    float s  = di * di;
    float4 hv = reinterpret_cast<const float4*>(h)[idx];
    float4 bv = reinterpret_cast<const float4*>(bias)[f4];
    float4 ov;
    ov.x = bv.x + s * hv.x;  ov.y = bv.y + s * hv.y;
    ov.z = bv.z + s * hv.z;  ov.w = bv.w + s * hv.w;
    reinterpret_cast<float4*>(out)[idx] = ov;
}

// ---------------- edge scatter: Y[dst] += dinv[src]*dinv[dst] * h[src] ----------------
// One wave per edge: single vector gather of h[src] row + coalesced f32 atomics.

template<int F>
__global__ __launch_bounds__(256) void edge_scatter_kernel(const float* __restrict__ h,
                                                           const float* __restrict__ dinv,
                                                           const int* __restrict__ ei,
                                                           float* __restrict__ out) {
    constexpr int VEC   = (F % 128 == 0) ? 4 : 2;   // 128 -> float4, 40 -> float2
    constexpr int LANES = F / VEC;                  // 32 or 20
    int e = blockIdx.x * (blockDim.x >> 5) + (threadIdx.x >> 5);
    if (e >= N_EDGES) return;
    int lane = threadIdx.x & 31;
    int s = ei[e];
    int d = ei[N_EDGES + e];
    float norm = dinv[s] * dinv[d];
    if (lane < LANES) {
        const float* __restrict__ hp = h   + (size_t)s * F + lane * VEC;
        float*       __restrict__ op = out + (size_t)d * F + lane * VEC;
        if constexpr (VEC == 4) {
            float4 v = *reinterpret_cast<const float4*>(hp);   // global_load_b128
            unsafeAtomicAdd(op + 0, norm * v.x);
            unsafeAtomicAdd(op + 1, norm * v.y);
            unsafeAtomicAdd(op + 2, norm * v.z);
            unsafeAtomicAdd(op + 3, norm * v.w);
        } else {
            float2 v = *reinterpret_cast<const float2*>(hp);   // global_load_b64
            unsafeAtomicAdd(op + 0, norm * v.x);
            unsafeAtomicAdd(op + 1, norm * v.y);
        }
    }
}

// ---------------- launch ----------------

extern "C" void kernel_launch(void* const* d_in, const int* in_sizes, int n_in,
                              void* d_out, int out_size, void* d_ws, size_t ws_size,
                              hipStream_t stream) {
    const float* x  = (const float*)d_in[0];
    const int*   ei = (const int*)d_in[1];     // [2, E]: row0 = src, row1 = dst
    const float* W1 = (const float*)d_in[2];
    const float* b1 = (const float*)d_in[3];
    const float* W2 = (const float*)d_in[4];
    const float* b2 = (const float*)d_in[5];
    const float* W3 = (const float*)d_in[6];
    const float* b3 = (const float*)d_in[7];
    float* out = (float*)d_out;

    // workspace: dinv | bufA (N*128) | bufB (N*128) | packed weights  ~103.5 MB
    char* ws = (char*)d_ws;
    float* dinv = (float*)ws;
    float* bufA = (float*)(ws + (((size_t)N_NODES * 4 + 255) & ~(size_t)255));
    float* bufB = bufA + (size_t)N_NODES * HID;
    v2f*   Wp1  = (v2f*)(bufB + (size_t)N_NODES * HID);
    const int NP1 = (F_IN / 4) * 8 * 32;   // 32000 v2f
    const int NP2 = (HID  / 4) * 8 * 32;   //  8192 v2f
    const int NP3 = (HID  / 4) * 3 * 32;   //  3072 v2f
    v2f* Wp2 = Wp1 + NP1;
    v2f* Wp3 = Wp2 + NP2;

    const int pairs   = N_NODES / 32;                 // 3125, exact
    const int gemmBlk = (pairs + 3) / 4;              // 4 waves/block, 32 rows/wave
    const int edgeBlk = (N_EDGES + 7) / 8;            // 8 waves/block, 1 edge/wave

    // pack weights into WMMA B-fragment layout (tiny, L2-resident)
    pack_w_kernel<8><<<(NP1 + 255) / 256, 256, 0, stream>>>(W1, Wp1, F_IN, HID);
    pack_w_kernel<8><<<(NP2 + 255) / 256, 256, 0, stream>>>(W2, Wp2, HID, HID);
    pack_w_kernel<3><<<(NP3 + 255) / 256, 256, 0, stream>>>(W3, Wp3, HID, N_CLS);

    // degrees + symmetric normalization
    deg_init_kernel <<<(N_NODES + 255) / 256, 256, 0, stream>>>(dinv);
    deg_count_kernel<<<(N_EDGES + 255) / 256, 256, 0, stream>>>(ei + N_EDGES, dinv);
    dinv_kernel     <<<(N_NODES + 255) / 256, 256, 0, stream>>>(dinv);

    // layer 1: h1 = x @ W1 ; agg1 = b1 + dinv^2*h1 + scatter
    gemm_wmma_kernel<8, HID, false><<<gemmBlk, 128, 0, stream>>>(x, Wp1, bufA, N_NODES, F_IN);
    self_bias_kernel<HID><<<((long)N_NODES * (HID/4) + 255) / 256, 256, 0, stream>>>(bufA, dinv, b1, bufB, N_NODES);
    edge_scatter_kernel<HID><<<edgeBlk, 256, 0, stream>>>(bufA, dinv, ei, bufB);

    // layer 2: h2 = relu(agg1) @ W2 ; agg2
    gemm_wmma_kernel<8, HID, true><<<gemmBlk, 128, 0, stream>>>(bufB, Wp2, bufA, N_NODES, HID);
    self_bias_kernel<HID><<<((long)N_NODES * (HID/4) + 255) / 256, 256, 0, stream>>>(bufA, dinv, b2, bufB, N_NODES);
    edge_scatter_kernel<HID><<<edgeBlk, 256, 0, stream>>>(bufA, dinv, ei, bufB);

    // layer 3: h3 = relu(agg2) @ W3 ; out = b3 + dinv^2*h3 + scatter (no relu)
    gemm_wmma_kernel<3, N_CLS, true><<<gemmBlk, 128, 0, stream>>>(bufB, Wp3, bufA, N_NODES, HID);
    self_bias_kernel<N_CLS><<<((long)N_NODES * (N_CLS/4) + 255) / 256, 256, 0, stream>>>(bufA, dinv, b3, out, N_NODES);
    edge_scatter_kernel<N_CLS><<<edgeBlk, 256, 0, stream>>>(bufA, dinv, ei, out);
}